// TraditionalGNN_6760278523984
// MI455X (gfx1250) — compile-verified
//
#include <hip/hip_runtime.h>

#define N_NODES 100000
#define N_EDGES 1600000
#define IN_DIM  128
#define HID     64

typedef float v2f __attribute__((ext_vector_type(2)));
typedef float v8f __attribute__((ext_vector_type(8)));

// ---------------------------------------------------------------------------
// WMMA f32 GEMM with LDS-staged B:
//   C[M,HID] = op(A[M,K] @ B[K,HID] (+bias) (ReLU?))
// Block = 128 threads (4 waves). Each block covers 16*TILES rows; wave w owns
// the 16-wide column tile [16w, 16w+16). B is staged once per block into LDS,
// pair-packed: ldsB[p*HID + c] = (B[2p][c], B[2p+1][c]) so each lane's B
// operand for V_WMMA_F32_16X16X4_F32 is one ds_load_b64.
//
// WMMA operand layouts (wave32, 16x16x4 f32):
//   A (16x4, 2 VGPR): lanes 0-15 row=lane {K=0,1}; lanes 16-31 row=lane-16 {K=2,3}
//   B (4x16, 2 VGPR): lanes 0-15 col=lane {K=0,1}; lanes 16-31 col=lane-16 {K=2,3}
//   C/D (16x16, 8 VGPR): VGPR v -> row v (lanes 0-15) / row v+8 (lanes 16-31),
//                        col = lane & 15
// ---------------------------------------------------------------------------
template <int K, int TILES, bool RELU>
__global__ __launch_bounds__(128)
void gemm_wmma_f32_lds(const float* __restrict__ A, const float* __restrict__ B,
                       const float* __restrict__ bias, float* __restrict__ C) {
  __shared__ float2 ldsB[(K / 2) * HID];

  const int tid     = threadIdx.x;
  const int lane    = tid & 31;
  const int wave    = tid >> 5;
  const int r       = lane & 15;
  const int khalf   = (lane >> 4) << 1;  // 0 or 2
  const int colBase = wave << 4;
  const int blockRow0 = blockIdx.x * (16 * TILES);

  // Stage B into LDS (pair-packed along K). Coalesced global reads,
  // conflict-free b64 LDS writes. Happens once per block.
  for (int i = tid; i < (K / 2) * HID; i += 128) {
    const int kp  = i / HID;
    const int col = i - kp * HID;
    float2 v;
    v.x = B[(size_t)(2 * kp) * HID + col];
    v.y = B[(size_t)(2 * kp + 1) * HID + col];
    ldsB[i] = v;
  }
  __syncthreads();

  const float bv = bias ? bias[colBase + r] : 0.0f;

  for (int m = 0; m < TILES; ++m) {
    const int row0 = blockRow0 + (m << 4);
    v8f acc = {};
    const float*  ap = A + (size_t)(row0 + r) * K + khalf;
    const float2* bp = ldsB + (khalf >> 1) * HID + colBase + r;
#pragma unroll 4
    for (int k0 = 0; k0 < K; k0 += 4) {
      v2f a, b;
      a.x = ap[0];
      a.y = ap[1];
      const float2 bb = *bp;
      b.x = bb.x;
      b.y = bb.y;
      ap += 4;
      bp += 2 * HID;
      // (neg_a, A, neg_b, B, c_mod, C, reuse_a, reuse_b)
      acc = __builtin_amdgcn_wmma_f32_16x16x4_f32(false, a, false, b,
                                                  (short)0, acc, false, false);
    }
#pragma unroll
    for (int v = 0; v < 8; ++v) {
      const int mm = (lane < 16) ? v : (v + 8);
      float val = acc[v] + bv;
      if (RELU) val = fmaxf(val, 0.0f);
      C[(size_t)(row0 + mm) * HID + colBase + r] = val;
    }
  }
}

// deg[i] = 1.0 (self-loop contribution)
__global__ void deg_init_kernel(float* __restrict__ deg, int n) {
  int i = blockIdx.x * blockDim.x + threadIdx.x;
  if (i < n) deg[i] = 1.0f;
}

// deg[dst[e]] += 1 for every edge
__global__ void deg_accum_kernel(const int* __restrict__ dst,
                                 float* __restrict__ deg, int e_count) {
  int e = blockIdx.x * blockDim.x + threadIdx.x;
  if (e < e_count) atomicAdd(&deg[dst[e]], 1.0f);
}

// deg -> rsqrt(deg) in place (becomes dinv)
__global__ void deg_rsqrt_kernel(float* __restrict__ deg, int n) {
  int i = blockIdx.x * blockDim.x + threadIdx.x;
  if (i < n) deg[i] = rsqrtf(deg[i]);
}

// agg[i, :] = hw[i, :] * dinv[i]^2   (self-loop edge i->i, norm = dinv^2)
__global__ void self_loop_kernel(const float* __restrict__ hw,
                                 const float* __restrict__ dinv,
                                 float* __restrict__ agg, int n) {
  int t = blockIdx.x * blockDim.x + threadIdx.x;
  if (t >= n * HID) return;
  int i = t >> 6;  // HID == 64
  float di = dinv[i];
  agg[t] = hw[t] * (di * di);
}

// Edge scatter: 16 lanes per edge, float4 per lane (coalesced 256B gather),
// 4 f32 atomics per lane into agg[dst] (L2-resident: hw+agg = 51MB << 192MB).
__global__ void edge_scatter_kernel(const float* __restrict__ hw,
                                    const float* __restrict__ dinv,
                                    const int* __restrict__ src,
                                    const int* __restrict__ dst,
                                    float* __restrict__ agg, int e_count) {
  int t = blockIdx.x * blockDim.x + threadIdx.x;
  int e = t >> 4;
  if (e >= e_count) return;
  int c = (t & 15) << 2;  // feature offset 0..60
  int s = src[e];
  int d = dst[e];
  float norm = dinv[s] * dinv[d];
  const float4 v = *(const float4*)(hw + (size_t)s * HID + c);
  float* o = agg + (size_t)d * HID + c;
  atomicAdd(o + 0, v.x * norm);
  atomicAdd(o + 1, v.y * norm);
  atomicAdd(o + 2, v.z * norm);
  atomicAdd(o + 3, v.w * norm);
}

// out[i] = b_out + sum_d (agg[i,d] + b_g[d]) * W_out[d]
__global__ void out_proj_kernel(const float* __restrict__ agg,
                                const float* __restrict__ b_g,
                                const float* __restrict__ W_out,
                                const float* __restrict__ b_out,
                                float* __restrict__ out, int n) {
  int i = blockIdx.x * blockDim.x + threadIdx.x;
  if (i >= n) return;
  const float* row = agg + (size_t)i * HID;
  float s = b_out[0];
#pragma unroll
  for (int d = 0; d < HID; ++d) s += (row[d] + b_g[d]) * W_out[d];
  out[i] = s;
}

extern "C" void kernel_launch(void* const* d_in, const int* in_sizes, int n_in,
                              void* d_out, int out_size, void* d_ws,
                              size_t ws_size, hipStream_t stream) {
  const float* x     = (const float*)d_in[0];
  const int*   eidx  = (const int*)d_in[1];
  const float* W_in  = (const float*)d_in[2];
  const float* b_in  = (const float*)d_in[3];
  const float* W_g   = (const float*)d_in[4];
  const float* b_g   = (const float*)d_in[5];
  const float* W_out = (const float*)d_in[6];
  const float* b_out = (const float*)d_in[7];
  float* out = (float*)d_out;

  const int N = N_NODES;
  const int E = in_sizes[1] / 2;

  // Workspace layout (floats): [h / agg : N*HID][hw : N*HID][deg/dinv : N]
  float* h   = (float*)d_ws;  // later reused as agg
  float* hw  = h + (size_t)N * HID;
  float* deg = hw + (size_t)N * HID;

  const int* src = eidx;      // edge_index[0, :]
  const int* dst = eidx + E;  // edge_index[1, :]

  // 1) h = relu(x @ W_in + b_in)   [WMMA f32 16x16x4, B in LDS, 160 rows/blk]
  gemm_wmma_f32_lds<IN_DIM, 10, true>
      <<<N / 160, 128, 0, stream>>>(x, W_in, b_in, h);

  // 2) hw = h @ W_g                [WMMA f32 16x16x4, B in LDS]
  gemm_wmma_f32_lds<HID, 10, false>
      <<<N / 160, 128, 0, stream>>>(h, W_g, nullptr, hw);

  // 3) degrees (self-loop => init 1), then dinv = rsqrt(deg)
  deg_init_kernel<<<(N + 255) / 256, 256, 0, stream>>>(deg, N);
  deg_accum_kernel<<<(E + 255) / 256, 256, 0, stream>>>(dst, deg, E);
  deg_rsqrt_kernel<<<(N + 255) / 256, 256, 0, stream>>>(deg, N);

  // 4) agg = self-loop term (overwrites h, which is dead now)
  self_loop_kernel<<<(N * HID + 255) / 256, 256, 0, stream>>>(hw, deg, h, N);

  // 5) scatter-add over edges (L2-resident atomics)
  edge_scatter_kernel<<<((unsigned)E * 16 + 255) / 256, 256, 0, stream>>>(
      hw, deg, src, dst, h, E);

  // 6) out = (agg + b_g) @ W_out + b_out
  out_proj_kernel<<<(N + 255) / 256, 256, 0, stream>>>(h, b_g, W_out, b_out,
                                                       out, N);
}